// Attention_19920058319433
// MI455X (gfx1250) — compile-verified
//
#include <hip/hip_runtime.h>
#include <math.h>

typedef __attribute__((ext_vector_type(2))) float v2f;
typedef __attribute__((ext_vector_type(8))) float v8f;

// Problem sizes (fixed by the reference)
constexpr int B = 64;
constexpr int S = 2048;
constexpr int H = 1024;

constexpr int CHUNKS = 8;                     // S-chunks per batch (blocks per batch)
constexpr int ROWS_PER_BLOCK = S / CHUNKS;    // 256
constexpr int TILE = 16;                      // rows per WMMA tile
constexpr int TILES = ROWS_PER_BLOCK / TILE;  // 16
constexpr int NWAVES = 8;                     // 256 threads / wave32
constexpr int KSLICE = H / NWAVES;            // 128 K-values per wave
constexpr int KSTEPS = KSLICE / 4;            // 32 wmma steps per wave per tile

// ---------------------------------------------------------------------------
// Kernel 1: per (b, chunk): online-softmax streaming pass.
//   - energies via V_WMMA_F32_16X16X4_F32 (16 rows at a time)
//   - raw tanh energies written to the weights output region
//   - partial (m, l, acc[H]) written to workspace
// ---------------------------------------------------------------------------
__global__ __launch_bounds__(256) void k_pass1(
    const float* __restrict__ x,     // [B,S,H]
    const float* __restrict__ w,     // [H]
    const float* __restrict__ bias,  // [1]
    float* __restrict__ e_out,       // [B,S]  (weights region of d_out; raw energies)
    float* __restrict__ pacc,        // [B*CHUNKS, H]
    float* __restrict__ pml)         // [B*CHUNKS, 2]  (m, l)
{
    __shared__ float w_s[H];          // 4 KB
    __shared__ float eprt[NWAVES][TILE];

    const int tid  = threadIdx.x;
    const int blk  = blockIdx.x;           // 0 .. B*CHUNKS-1
    const int b    = blk / CHUNKS;
    const int c    = blk % CHUNKS;
    const int sbeg = c * ROWS_PER_BLOCK;

    // stage w into LDS (B-matrix source for WMMA)
    for (int i = tid; i < H; i += 256) w_s[i] = w[i];
    __syncthreads();

    const float bias_v = bias[0];

    const int wave  = tid >> 5;
    const int lane  = tid & 31;
    const int row   = lane & 15;           // A-matrix: lanes 0-15 and 16-31 both hold M=0..15
    const int koff  = (lane >> 4) << 1;    // lanes 16-31 hold K+2, K+3
    const int kbase = wave * KSLICE;

    float  m_run = -INFINITY;
    float  l_run = 0.0f;
    float  acc0 = 0.f, acc1 = 0.f, acc2 = 0.f, acc3 = 0.f;
    const int h0 = tid * 4;                // each thread owns 4 h-values

    const float* __restrict__ xb = x + (size_t)b * S * H;

    for (int t = 0; t < TILES; ++t) {
        const int s0 = sbeg + t * TILE;

        // ---- 16 row-dots via WMMA over this wave's K slice -----------------
        v8f cacc;
        #pragma unroll
        for (int i = 0; i < 8; ++i) cacc[i] = 0.0f;

        const float* __restrict__ arow = xb + (size_t)(s0 + row) * H + kbase + koff;
        const float* __restrict__ wrow = &w_s[kbase + koff];

        #pragma unroll 4
        for (int it = 0; it < KSTEPS; ++it) {
            v2f a  = *(const v2f*)(arow + it * 4);   // global_load_b64, per-lane row
            v2f bb = *(const v2f*)(wrow + it * 4);   // ds_load_b64, w replicated in all N cols
            cacc = __builtin_amdgcn_wmma_f32_16x16x4_f32(
                false, a, false, bb, (short)0, cacc, false, false);
        }

        // D[m][n] identical for all n. C/D layout: vgpr r, lanes 0-15 -> M=r,
        // lanes 16-31 -> M=r+8. Publish this wave's 16 K-partial dots.
        if (lane == 0) {
            #pragma unroll
            for (int i = 0; i < 8; ++i) eprt[wave][i] = cacc[i];
        }
        if (lane == 16) {
            #pragma unroll
            for (int i = 0; i < 8; ++i) eprt[wave][8 + i] = cacc[i];
        }
        __syncthreads();

        // ---- every thread reduces partials -> 16 energies (identical) ------
        float e[TILE];
        #pragma unroll
        for (int j = 0; j < TILE; ++j) {
            float s = bias_v;
            #pragma unroll
            for (int wv = 0; wv < NWAVES; ++wv) s += eprt[wv][j];
            e[j] = tanhf(s);
            if (tid == j) e_out[(size_t)b * S + s0 + j] = e[j];  // raw energy
        }
        __syncthreads();   // eprt reused next tile

        // ---- online softmax update ----------------------------------------
        float m_new = m_run;
        #pragma unroll
        for (int j = 0; j < TILE; ++j) m_new = fmaxf(m_new, e[j]);
        const float scale = __expf(m_run - m_new);
        float p[TILE];
        float psum = 0.0f;
        #pragma unroll
        for (int j = 0; j < TILE; ++j) { p[j] = __expf(e[j] - m_new); psum += p[j]; }
        l_run = l_run * scale + psum;
        m_run = m_new;

        acc0 *= scale; acc1 *= scale; acc2 *= scale; acc3 *= scale;
        // re-read the just-streamed tile (WGP$/L2-hot), coalesced b128
        #pragma unroll 4
        for (int j = 0; j < TILE; ++j) {
            const float4 xv = *(const float4*)(xb + (size_t)(s0 + j) * H + h0);
            acc0 += p[j] * xv.x;
            acc1 += p[j] * xv.y;
            acc2 += p[j] * xv.z;
            acc3 += p[j] * xv.w;
        }
    }

    // ---- write chunk partials ---------------------------------------------
    float4 out; out.x = acc0; out.y = acc1; out.z = acc2; out.w = acc3;
    *(float4*)(pacc + (size_t)blk * H + h0) = out;
    if (tid == 0) { pml[blk * 2] = m_run; pml[blk * 2 + 1] = l_run; }
}

// ---------------------------------------------------------------------------
// Kernel 2: combine CHUNKS partials per batch -> context, store (M, L)
// ---------------------------------------------------------------------------
__global__ __launch_bounds__(256) void k_combine(
    const float* __restrict__ pacc,  // [B*CHUNKS, H]
    const float* __restrict__ pml,   // [B*CHUNKS, 2]
    float* __restrict__ ctx,         // [B, H]
    float* __restrict__ ML)          // [B, 2]
{
    const int b   = blockIdx.x;
    const int tid = threadIdx.x;

    float m[CHUNKS], l[CHUNKS];
    float M = -INFINITY;
    #pragma unroll
    for (int c = 0; c < CHUNKS; ++c) {
        m[c] = pml[(b * CHUNKS + c) * 2];
        l[c] = pml[(b * CHUNKS + c) * 2 + 1];
        M = fmaxf(M, m[c]);
    }
    float f[CHUNKS];
    float L = 0.0f;
    #pragma unroll
    for (int c = 0; c < CHUNKS; ++c) { f[c] = __expf(m[c] - M); L += l[c] * f[c]; }
    const float invL = 1.0f / L;

    const int h0 = tid * 4;
    float s0 = 0.f, s1 = 0.f, s2 = 0.f, s3 = 0.f;
    #pragma unroll
    for (int c = 0; c < CHUNKS; ++c) {
        const float4 v = *(const float4*)(pacc + (size_t)(b * CHUNKS + c) * H + h0);
        s0 += f[c] * v.x; s1 += f[c] * v.y; s2 += f[c] * v.z; s3 += f[c] * v.w;
    }
    float4 o; o.x = s0 * invL; o.y = s1 * invL; o.z = s2 * invL; o.w = s3 * invL;
    *(float4*)(ctx + (size_t)b * H + h0) = o;

    if (tid == 0) { ML[b * 2] = M; ML[b * 2 + 1] = L; }
}

// ---------------------------------------------------------------------------
// Kernel 3: weights = exp(e - M)/L in place in the weights output region
// ---------------------------------------------------------------------------
__global__ __launch_bounds__(256) void k_weights(
    float* __restrict__ wgt,        // [B,S] raw energies -> softmax weights
    const float* __restrict__ ML)   // [B, 2]
{
    const int i = blockIdx.x * 256 + threadIdx.x;   // < B*S
    const int b = i / S;
    const float M = ML[b * 2];
    const float L = ML[b * 2 + 1];
    wgt[i] = __expf(wgt[i] - M) / L;
}

// ---------------------------------------------------------------------------
extern "C" void kernel_launch(void* const* d_in, const int* in_sizes, int n_in,
                              void* d_out, int out_size, void* d_ws, size_t ws_size,
                              hipStream_t stream)
{
    const float* x    = (const float*)d_in[0];   // [B,S,H]
    const float* w    = (const float*)d_in[1];   // [H]
    const float* bias = (const float*)d_in[2];   // [1]

    float* ctx = (float*)d_out;                    // [B,H]
    float* wgt = (float*)d_out + (size_t)B * H;    // [B,S]

    float* pacc = (float*)d_ws;                              // B*CHUNKS*H floats
    float* pml  = pacc + (size_t)B * CHUNKS * H;             // B*CHUNKS*2 floats
    float* ML   = pml + (size_t)B * CHUNKS * 2;              // B*2 floats

    k_pass1  <<<B * CHUNKS, 256, 0, stream>>>(x, w, bias, wgt, pacc, pml);
    k_combine<<<B,          256, 0, stream>>>(pacc, pml, ctx, ML);
    k_weights<<<(B * S) / 256, 256, 0, stream>>>(wgt, ML);
}